// NN_13907104104967
// MI455X (gfx1250) — compile-verified
//
#include <hip/hip_runtime.h>

typedef __attribute__((ext_vector_type(2))) float v2f;
typedef __attribute__((ext_vector_type(8))) float v8f;

// D = A(16x4,f32) x B(4x16,f32) + C(16x16,f32), wave32, exact fp32.
__device__ __forceinline__ v8f wmma_f32(v2f a, v2f b, v8f c) {
    return __builtin_amdgcn_wmma_f32_16x16x4_f32(
        /*neg_a=*/false, a, /*neg_b=*/false, b,
        /*c_mod=*/(short)0, c, /*reuse_a=*/false, /*reuse_b=*/false);
}

// C[B x N] = post( (A .* rowscale?) [B x K] @ W[N x K]^T + bias? ) .* mask?
// One wave -> 32(M) x 64(N) tile (2x4 WMMA subtiles); 4 waves/block -> 128 rows.
// - Out-of-range N columns: W row pointer clamped; garbage stays in its own D
//   column (WMMA mixes only over K) and is never stored -> divergence-free loop.
// - Software-pipelined: iteration i+1's six fragments are loaded into a second
//   register set before iteration i's 8 WMMAs, so global-load latency is hidden
//   behind a full WMMA burst instead of being exposed per fragment.
template <bool HAS_SCALE>
__global__ __launch_bounds__(128) void gemm32_wmma(
    const float* __restrict__ A, const float* __restrict__ rowscale,
    const float* __restrict__ W, const float* __restrict__ bias,
    const float* __restrict__ mask, float* __restrict__ C,
    int K, int N, int do_relu)
{
    const int wave = threadIdx.x >> 5;
    const int lane = threadIdx.x & 31;
    const int half = lane >> 4;      // 0: K pair {0,1}, 1: K pair {2,3}
    const int l    = lane & 15;      // M (for A) or N (for B/C/D) within tile
    const int m0   = (blockIdx.y * 4 + wave) * 32;
    const int n0   = blockIdx.x * 64;

    const int r0 = m0 + l, r1 = m0 + 16 + l;
    const float s0 = HAS_SCALE ? rowscale[r0] : 1.0f;
    const float s1 = HAS_SCALE ? rowscale[r1] : 1.0f;
    const float* __restrict__ A0 = A + (size_t)r0 * K + (half << 1);
    const float* __restrict__ A1 = A + (size_t)r1 * K + (half << 1);

    const int c0 = n0 + l;
    const int cc0 = (c0      < N) ? c0      : (N - 1);
    const int cc1 = (c0 + 16 < N) ? c0 + 16 : (N - 1);
    const int cc2 = (c0 + 32 < N) ? c0 + 32 : (N - 1);
    const int cc3 = (c0 + 48 < N) ? c0 + 48 : (N - 1);
    const float* __restrict__ B0 = W + (size_t)cc0 * K + (half << 1);
    const float* __restrict__ B1 = W + (size_t)cc1 * K + (half << 1);
    const float* __restrict__ B2 = W + (size_t)cc2 * K + (half << 1);
    const float* __restrict__ B3 = W + (size_t)cc3 * K + (half << 1);

    v8f acc[2][4] = {};

    // Prologue: load k-step 0.
    v2f a0, a1, b0, b1, b2, b3;
    a0.x = A0[0];  a0.y = A0[1];
    a1.x = A1[0];  a1.y = A1[1];
    b0.x = B0[0];  b0.y = B0[1];
    b1.x = B1[0];  b1.y = B1[1];
    b2.x = B2[0];  b2.y = B2[1];
    b3.x = B3[0];  b3.y = B3[1];

    #pragma unroll 2
    for (int k0 = 4; k0 < K; k0 += 4) {
        // Issue next k-step's loads first (independent of current WMMAs).
        v2f na0, na1, nb0, nb1, nb2, nb3;
        na0.x = A0[k0];  na0.y = A0[k0 + 1];
        na1.x = A1[k0];  na1.y = A1[k0 + 1];
        nb0.x = B0[k0];  nb0.y = B0[k0 + 1];
        nb1.x = B1[k0];  nb1.y = B1[k0 + 1];
        nb2.x = B2[k0];  nb2.y = B2[k0 + 1];
        nb3.x = B3[k0];  nb3.y = B3[k0 + 1];

        if (HAS_SCALE) {
            a0.x *= s0;  a0.y *= s0;
            a1.x *= s1;  a1.y *= s1;
        }
        acc[0][0] = wmma_f32(a0, b0, acc[0][0]);
        acc[1][0] = wmma_f32(a1, b0, acc[1][0]);
        acc[0][1] = wmma_f32(a0, b1, acc[0][1]);
        acc[1][1] = wmma_f32(a1, b1, acc[1][1]);
        acc[0][2] = wmma_f32(a0, b2, acc[0][2]);
        acc[1][2] = wmma_f32(a1, b2, acc[1][2]);
        acc[0][3] = wmma_f32(a0, b3, acc[0][3]);
        acc[1][3] = wmma_f32(a1, b3, acc[1][3]);

        a0 = na0;  a1 = na1;
        b0 = nb0;  b1 = nb1;  b2 = nb2;  b3 = nb3;
    }

    // Epilogue k-step.
    if (HAS_SCALE) {
        a0.x *= s0;  a0.y *= s0;
        a1.x *= s1;  a1.y *= s1;
    }
    acc[0][0] = wmma_f32(a0, b0, acc[0][0]);
    acc[1][0] = wmma_f32(a1, b0, acc[1][0]);
    acc[0][1] = wmma_f32(a0, b1, acc[0][1]);
    acc[1][1] = wmma_f32(a1, b1, acc[1][1]);
    acc[0][2] = wmma_f32(a0, b2, acc[0][2]);
    acc[1][2] = wmma_f32(a1, b2, acc[1][2]);
    acc[0][3] = wmma_f32(a0, b3, acc[0][3]);
    acc[1][3] = wmma_f32(a1, b3, acc[1][3]);

    #pragma unroll
    for (int j = 0; j < 4; ++j) {
        const int col = n0 + 16 * j + l;
        if (col < N) {
            const float bv = bias ? bias[col] : 0.0f;
            const float mv = mask ? mask[col] : 1.0f;
            #pragma unroll
            for (int t = 0; t < 2; ++t) {
                #pragma unroll
                for (int r = 0; r < 8; ++r) {
                    const int row = m0 + t * 16 + r + (half << 3); // VGPR r -> M=r / r+8
                    float v = acc[t][j][r] + bv;
                    if (do_relu) v = fmaxf(v, 0.0f);
                    C[(size_t)row * N + col] = v * mv;
                }
            }
        }
    }
}

// Per-row 1/||row|| and (rowsum)/||row||. One block per row.
__global__ __launch_bounds__(128) void rowstats(
    const float* __restrict__ A, int K,
    float* __restrict__ invnorm, float* __restrict__ snorm)
{
    const int b = blockIdx.x, tid = threadIdx.x;
    const float* row = A + (size_t)b * K;
    float ssq = 0.0f, s = 0.0f;
    for (int k = tid; k < K; k += 128) {
        float v = row[k];
        ssq = fmaf(v, v, ssq);
        s  += v;
    }
    __shared__ float sh1[128], sh2[128];
    sh1[tid] = ssq; sh2[tid] = s;
    __syncthreads();
    for (int st = 64; st; st >>= 1) {
        if (tid < st) { sh1[tid] += sh1[tid + st]; sh2[tid] += sh2[tid + st]; }
        __syncthreads();
    }
    if (tid == 0) {
        const float inv = 1.0f / sqrtf(sh1[0]);
        invnorm[b] = inv;
        if (snorm) snorm[b] = sh2[0] * inv;
    }
}

// Deterministic per-chunk partials: colpart[c][n] = sum_b yn[b,n]*sn[b],
// sqpart[c] = sum_{b,n} yn[b,n]^2.  blockDim == N (power of two <= 256).
__global__ void coldot_partial(
    const float* __restrict__ Y, const float* __restrict__ invyn,
    const float* __restrict__ sn, float* __restrict__ colpart,
    float* __restrict__ sqpart, int Brows, int N, int chunk)
{
    const int n  = threadIdx.x;
    const int b0 = blockIdx.x * chunk;
    int b1 = b0 + chunk; if (b1 > Brows) b1 = Brows;
    float col = 0.0f, sq = 0.0f;
    for (int b = b0; b < b1; ++b) {
        const float v = Y[(size_t)b * N + n] * invyn[b];
        col = fmaf(v, sn[b], col);
        sq  = fmaf(v, v, sq);
    }
    colpart[(size_t)blockIdx.x * N + n] = col;
    __shared__ float sh[256];
    sh[n] = sq;
    __syncthreads();
    for (int st = N >> 1; st; st >>= 1) {
        if (n < st) sh[n] += sh[n + st];
        __syncthreads();
    }
    if (n == 0) sqpart[blockIdx.x] = sh[0];
}

// score[n] = (1 - theta)*rowsum(H)[n] + coldot[n]; min-max; threshold 0.5.
__global__ void finalize_mask(
    const float* __restrict__ H, const float* __restrict__ colpart,
    const float* __restrict__ sqpart, int nchunks, int N, int K, int Brows,
    float* __restrict__ hm)
{
    const int n = threadIdx.x;
    float coldot = 0.0f;
    for (int c = 0; c < nchunks; ++c) coldot += colpart[(size_t)c * N + n];
    float rowsum = 0.0f;
    const float* hrow = H + (size_t)n * K;
    for (int k = 0; k < K; ++k) rowsum += hrow[k];

    __shared__ float shtheta;
    if (n == 0) {
        float t = 0.0f;
        for (int c = 0; c < nchunks; ++c) t += sqpart[c];
        shtheta = t / ((float)Brows * (float)N);
    }
    __syncthreads();
    const float score = (1.0f - shtheta) * rowsum + coldot;   // LR = 1

    __shared__ float shmin[256], shmax[256];
    shmin[n] = score; shmax[n] = score;
    __syncthreads();
    for (int st = N >> 1; st; st >>= 1) {
        if (n < st) {
            shmin[n] = fminf(shmin[n], shmin[n + st]);
            shmax[n] = fmaxf(shmax[n], shmax[n + st]);
        }
        __syncthreads();
    }
    const float nv = (score - shmin[0]) / (shmax[0] - shmin[0] + 1e-8f);
    hm[n] = (nv < 0.5f) ? 0.0f : 1.0f;
}

extern "C" void kernel_launch(void* const* d_in, const int* in_sizes, int n_in,
                              void* d_out, int out_size, void* d_ws, size_t ws_size,
                              hipStream_t stream)
{
    (void)in_sizes; (void)n_in; (void)out_size; (void)ws_size;

    const float* x  = (const float*)d_in[0];
    const float* m0 = (const float*)d_in[1];
    const float* m1 = (const float*)d_in[2];
    const float* m2 = (const float*)d_in[3];
    const float* W1 = (const float*)d_in[4];
    const float* b1 = (const float*)d_in[5];
    const float* W2 = (const float*)d_in[6];
    const float* b2 = (const float*)d_in[7];
    const float* W3 = (const float*)d_in[8];
    const float* b3 = (const float*)d_in[9];
    const float* W4 = (const float*)d_in[10];
    const float* b4 = (const float*)d_in[11];
    const float* H1 = (const float*)d_in[12];
    const float* H2 = (const float*)d_in[13];
    const float* H3 = (const float*)d_in[14];

    const int B = 8192, DIN = 1024, DOUT = 1000;
    const int h0 = 256, h1 = 128, h2 = 64;

    float* out = (float*)d_out;
    float* hm0 = out + (size_t)B * DOUT;
    float* hm1 = hm0 + h0;
    float* hm2 = hm1 + h1;

    float* ws = (float*)d_ws;
    size_t off = 0;
    float* x1 = ws + off; off += (size_t)B * h0;
    float* x2 = ws + off; off += (size_t)B * h1;
    float* x3 = ws + off; off += (size_t)B * h2;
    float* Y  = ws + off; off += (size_t)B * h0;
    float* invrn = ws + off; off += B;
    float* sn    = ws + off; off += B;
    float* invyn = ws + off; off += B;
    const int CHUNK = 128, NCH = B / CHUNK;   // 64 chunks
    float* colpart = ws + off; off += (size_t)NCH * h0;
    float* sqpart  = ws + off; off += NCH;

    const dim3 blk(128);
    const int gy = B / 128;   // 4 waves/block * 32 rows/wave

    // ---- Layer 1: hebbian(x, H1) -> hm0 ; x1 = relu(x@W1.T + b1)*m0 ----
    rowstats<<<B, 128, 0, stream>>>(x, DIN, invrn, sn);
    gemm32_wmma<true ><<<dim3(h0 / 64, gy), blk, 0, stream>>>(x, invrn, H1, nullptr, nullptr, Y, DIN, h0, 0);
    rowstats<<<B, 128, 0, stream>>>(Y, h0, invyn, nullptr);
    coldot_partial<<<NCH, h0, 0, stream>>>(Y, invyn, sn, colpart, sqpart, B, h0, CHUNK);
    finalize_mask<<<1, h0, 0, stream>>>(H1, colpart, sqpart, NCH, h0, DIN, B, hm0);
    gemm32_wmma<false><<<dim3(h0 / 64, gy), blk, 0, stream>>>(x, nullptr, W1, b1, m0, x1, DIN, h0, 1);

    // ---- Layer 2 ----
    rowstats<<<B, 128, 0, stream>>>(x1, h0, invrn, sn);
    gemm32_wmma<true ><<<dim3(h1 / 64, gy), blk, 0, stream>>>(x1, invrn, H2, nullptr, nullptr, Y, h0, h1, 0);
    rowstats<<<B, 128, 0, stream>>>(Y, h1, invyn, nullptr);
    coldot_partial<<<NCH, h1, 0, stream>>>(Y, invyn, sn, colpart, sqpart, B, h1, CHUNK);
    finalize_mask<<<1, h1, 0, stream>>>(H2, colpart, sqpart, NCH, h1, h0, B, hm1);
    gemm32_wmma<false><<<dim3(h1 / 64, gy), blk, 0, stream>>>(x1, nullptr, W2, b2, m1, x2, h0, h1, 1);

    // ---- Layer 3 ----
    rowstats<<<B, 128, 0, stream>>>(x2, h1, invrn, sn);
    gemm32_wmma<true ><<<dim3(h2 / 64, gy), blk, 0, stream>>>(x2, invrn, H3, nullptr, nullptr, Y, h1, h2, 0);
    rowstats<<<B, 128, 0, stream>>>(Y, h2, invyn, nullptr);
    coldot_partial<<<NCH, h2, 0, stream>>>(Y, invyn, sn, colpart, sqpart, B, h2, CHUNK);
    finalize_mask<<<1, h2, 0, stream>>>(H3, colpart, sqpart, NCH, h2, h1, B, hm2);
    gemm32_wmma<false><<<dim3(h2 / 64, gy), blk, 0, stream>>>(x2, nullptr, W3, b3, m2, x3, h1, h2, 1);

    // ---- Layer 4: out = relu(x3@W4.T + b4) ----
    gemm32_wmma<false><<<dim3((DOUT + 63) / 64, gy), blk, 0, stream>>>(x3, nullptr, W4, b4, nullptr, out, h2, DOUT, 1);
}